// GCN_delta_66872640799058
// MI455X (gfx1250) — compile-verified
//
#include <hip/hip_runtime.h>
#include <hip/hip_bf16.h>

#define N_NODES 100000
#define E_EDGES 1600000
#define F_IN    128
#define HID     128
#define C_OUT   64
#define BN_EPS  1e-5f

typedef __attribute__((ext_vector_type(2))) float v2f;
typedef __attribute__((ext_vector_type(8))) float v8f;

// ---------------------------------------------------------------------------
// C[M,Nc] = A[M,K] @ B[K,Nc]  (all row-major fp32, M%16==0, K%4==0, Nc%16==0)
// One wave -> one 16x16 output tile via V_WMMA_F32_16X16X4_F32 (exact fp32).
// blockDim.x = (Nc/16)*32, gridDim.x = M/16.
// ---------------------------------------------------------------------------
__global__ void wmma_gemm_f32(const float* __restrict__ A,
                              const float* __restrict__ B,
                              float* __restrict__ Cm,
                              int M, int K, int Nc) {
    const int lane = threadIdx.x & 31;
    const int wave = threadIdx.x >> 5;
    const int row0 = blockIdx.x * 16;
    const int col0 = wave * 16;
    const int idx  = lane & 15;   // M index (A) / N index (B, C/D)
    const int grp  = lane >> 4;   // K sub-group 0/1

    v8f acc = {0.f, 0.f, 0.f, 0.f, 0.f, 0.f, 0.f, 0.f};

    // A: lane grp 0 reads (k0, k0+1), grp 1 reads (k0+2, k0+3) of row (row0+idx)
    const float* arow = A + (size_t)(row0 + idx) * K + 2 * grp;

    for (int k0 = 0; k0 < K; k0 += 4) {
        v2f a = *(const v2f*)(arow + k0);                  // 8B-aligned (even idx)
        const float* bp = B + (size_t)(k0 + 2 * grp) * Nc + col0 + idx;
        v2f b;
        b.x = bp[0];          // B[k0+2g    ][col0+idx]
        b.y = bp[Nc];         // B[k0+2g + 1][col0+idx]
        acc = __builtin_amdgcn_wmma_f32_16x16x4_f32(
            /*neg_a=*/false, a, /*neg_b=*/false, b,
            /*c_mod=*/(short)0, acc, /*reuse_a=*/false, /*reuse_b=*/false);
    }

    // D layout: lanes 0-15 -> M = v, lanes 16-31 -> M = v + 8
    float* crow = Cm + (size_t)(row0 + 8 * grp) * Nc + col0 + idx;
#pragma unroll
    for (int v = 0; v < 8; ++v)
        crow[(size_t)v * Nc] = acc[v];
}

// ---------------------------------------------------------------------------
// Edge aggregation: Out[dst] += X[src] * w   (Out pre-zeroed). Wave per edge.
// ---------------------------------------------------------------------------
template <int D> struct VecOf;
template <> struct VecOf<128> { using T = float4; };
template <> struct VecOf<64>  { using T = float2; };

template <int D>
__global__ void edge_aggregate(const float* __restrict__ X,
                               const int* __restrict__ src,
                               const int* __restrict__ dst,
                               const float* __restrict__ w,
                               float* __restrict__ Out,
                               int nEdges) {
    constexpr int PER = D / 32;
    using V = typename VecOf<D>::T;
    const int lane   = threadIdx.x & 31;
    const int waveId = blockIdx.x * (blockDim.x >> 5) + (threadIdx.x >> 5);
    const int nWaves = gridDim.x * (blockDim.x >> 5);

    for (int e = waveId; e < nEdges; e += nWaves) {
        const int   s  = src[e];
        const int   d  = dst[e];
        const float wt = w[e];
        const float* xp = X + (size_t)s * D + lane * PER;
        float*       op = Out + (size_t)d * D + lane * PER;
        V xv = *(const V*)xp;                        // b128 / b64 gather
        const float* xf = (const float*)&xv;
#pragma unroll
        for (int j = 0; j < PER; ++j)
            atomicAdd(op + j, xf[j] * wt);           // global_atomic_add_f32
    }
}

// ---------------------------------------------------------------------------
// h = relu( gamma * (h + b - mean) * rsqrt(var + eps) + beta )   in place
// ---------------------------------------------------------------------------
__global__ void bias_bn_relu(float* __restrict__ H,
                             const float* __restrict__ b,
                             const float* __restrict__ gamma,
                             const float* __restrict__ beta,
                             const float* __restrict__ mean,
                             const float* __restrict__ var,
                             int total) {
    int i = blockIdx.x * blockDim.x + threadIdx.x;
    if (i >= total) return;
    int f = i & (HID - 1);
    float h   = H[i] + b[f];
    float inv = rsqrtf(var[f] + BN_EPS);
    h = gamma[f] * (h - mean[f]) * inv + beta[f];
    H[i] = fmaxf(h, 0.f);
}

// ---------------------------------------------------------------------------
// Delta-merge + log_softmax. One wave per row, 2 cols/lane, wave32 butterflies.
// ---------------------------------------------------------------------------
__global__ void finalize_rows(const float* __restrict__ agg2,
                              const float* __restrict__ b2,
                              const float* __restrict__ prev,
                              const unsigned char* __restrict__ sens,
                              const unsigned char* __restrict__ insens,
                              float* __restrict__ logsm,
                              float* __restrict__ outBuf,
                              int n) {
    const int lane = threadIdx.x & 31;
    const int row  = blockIdx.x * (blockDim.x >> 5) + (threadIdx.x >> 5);
    if (row >= n) return;

    const bool s   = sens[row] != 0;
    const bool ins = insens[row] != 0;

    float o[2];
#pragma unroll
    for (int j = 0; j < 2; ++j) {
        const int f  = lane + 32 * j;
        const size_t ix = (size_t)row * C_OUT + f;
        float h2   = agg2[ix] + b2[f];
        float base = s ? h2 : prev[ix];               // keep = ~sens
        o[j] = base + (ins ? h2 : 0.f);
    }

    float m = fmaxf(o[0], o[1]);
#pragma unroll
    for (int off = 16; off >= 1; off >>= 1)
        m = fmaxf(m, __shfl_xor(m, off, 32));

    float sum = __expf(o[0] - m) + __expf(o[1] - m);
#pragma unroll
    for (int off = 16; off >= 1; off >>= 1)
        sum += __shfl_xor(sum, off, 32);
    const float lse = __logf(sum);

#pragma unroll
    for (int j = 0; j < 2; ++j) {
        const int f  = lane + 32 * j;
        const size_t ix = (size_t)row * C_OUT + f;
        logsm[ix]  = o[j] - m - lse;
        outBuf[ix] = o[j];
    }
}

// ---------------------------------------------------------------------------
extern "C" void kernel_launch(void* const* d_in, const int* in_sizes, int n_in,
                              void* d_out, int out_size, void* d_ws, size_t ws_size,
                              hipStream_t stream) {
    const float* features = (const float*)d_in[0];
    const int*   esrc     = (const int*)d_in[1];
    const int*   edst     = (const int*)d_in[2];
    const float* ew       = (const float*)d_in[3];
    const float* W1       = (const float*)d_in[4];
    const float* b1       = (const float*)d_in[5];
    const float* gamma1   = (const float*)d_in[6];
    const float* beta1    = (const float*)d_in[7];
    const float* mean1    = (const float*)d_in[8];
    const float* var1     = (const float*)d_in[9];
    const float* W2       = (const float*)d_in[10];
    const float* b2       = (const float*)d_in[11];
    const float* prev     = (const float*)d_in[12];
    const unsigned char* sens   = (const unsigned char*)d_in[13];
    const unsigned char* insens = (const unsigned char*)d_in[14];

    // Workspace layout (102.4 MB total, second layer aliases the first):
    //   xw1  = [N, HID]  X@W1
    //   h1   = [N, HID]  aggregated + BN/ReLU   (later reused as agg2 [N, C])
    //   hw2  = [N, C]    H@W2                   (aliases xw1)
    float* xw1  = (float*)d_ws;
    float* h1   = xw1 + (size_t)N_NODES * HID;
    float* hw2  = xw1;
    float* agg2 = h1;

    float* out_logsm = (float*)d_out;
    float* out_emb   = out_logsm + (size_t)N_NODES * C_OUT;

    const int rowTiles = N_NODES / 16;                 // 6250

    // ---- Layer 1 ----
    hipMemsetAsync(h1, 0, (size_t)N_NODES * HID * sizeof(float), stream);
    wmma_gemm_f32<<<rowTiles, (HID / 16) * 32, 0, stream>>>(
        features, W1, xw1, N_NODES, F_IN, HID);
    edge_aggregate<HID><<<4096, 256, 0, stream>>>(xw1, esrc, edst, ew, h1, E_EDGES);
    bias_bn_relu<<<(N_NODES * HID + 255) / 256, 256, 0, stream>>>(
        h1, b1, gamma1, beta1, mean1, var1, N_NODES * HID);

    // ---- Layer 2 ----
    wmma_gemm_f32<<<rowTiles, (C_OUT / 16) * 32, 0, stream>>>(
        h1, W2, hw2, N_NODES, HID, C_OUT);
    hipMemsetAsync(agg2, 0, (size_t)N_NODES * C_OUT * sizeof(float), stream);
    edge_aggregate<C_OUT><<<4096, 256, 0, stream>>>(hw2, esrc, edst, ew, agg2, E_EDGES);

    // ---- Merge + log_softmax ----
    finalize_rows<<<(N_NODES + 7) / 8, 256, 0, stream>>>(
        agg2, b2, prev, sens, insens, out_logsm, out_emb, N_NODES);
}